// GNN_11141145166538
// MI455X (gfx1250) — compile-verified
//
#include <hip/hip_runtime.h>

typedef float v2f __attribute__((ext_vector_type(2)));
typedef float v8f __attribute__((ext_vector_type(8)));

#define NN 50000
#define EE 800000
#define NEG_SLOPE 0.1f

__device__ __forceinline__ float leaky(float v) { return v > 0.f ? v : NEG_SLOPE * v; }

// D = A(16x4 f32) * B(4x16 f32) + C(16x16 f32), wave32
__device__ __forceinline__ v8f wmma4(v2f a, v2f b, v8f c) {
    return __builtin_amdgcn_wmma_f32_16x16x4_f32(false, a, false, b, (short)0, c, false, false);
}

// ---------------- degree / norm ----------------
__global__ void k_init_deg(float* __restrict__ deg) {
    int i = blockIdx.x * 256 + threadIdx.x;
    if (i < NN) deg[i] = 1.0f;   // self-loop counts once
}

__global__ void k_count_deg(const int* __restrict__ col, float* __restrict__ deg) {
    int e = blockIdx.x * 256 + threadIdx.x;
    if (e < EE) atomicAdd(&deg[col[e]], 1.0f);
}

__global__ void k_rsqrt(float* __restrict__ deg) {
    int i = blockIdx.x * 256 + threadIdx.x;
    if (i < NN) deg[i] = rsqrtf(deg[i]);   // deg >= 1 always
}

// dst[i][d] = dinv[i]^2 * src[i][d]   (self-loop term), D = 64
__global__ void k_self_init(const float* __restrict__ src, const float* __restrict__ dinv,
                            float* __restrict__ dst) {
    int idx = blockIdx.x * 256 + threadIdx.x;
    if (idx < NN * 64) {
        int i = idx >> 6;
        float di = dinv[i];
        dst[idx] = di * di * src[idx];
    }
}

// dst[col][d] += dinv[row]*dinv[col] * src[row][d], D = 64; one thread per (edge, feature)
__global__ void k_scatter64(const int* __restrict__ row, const int* __restrict__ col,
                            const float* __restrict__ dinv, const float* __restrict__ src,
                            float* __restrict__ dst) {
    int idx = blockIdx.x * 256 + threadIdx.x;   // EE*64 = 51.2M threads
    int e = idx >> 6;
    int d = idx & 63;
    int r = row[e];
    int c = col[e];
    float w = dinv[r] * dinv[c];
    atomicAdd(&dst[c * 64 + d], w * src[r * 64 + d]);
}

// h[i][d] = leaky(h[i][d] + b[d]), D = 64
__global__ void k_bias_act64(float* __restrict__ h, const float* __restrict__ b) {
    int idx = blockIdx.x * 256 + threadIdx.x;
    if (idx < NN * 64) h[idx] = leaky(h[idx] + b[idx & 63]);
}

// ---------------- WMMA GEMM: C[3125*16 x NO] = A[.. x K] * B[K x NO] ----------------
// blockDim = 32 * (NO/16); wave w computes column tile w of row tile blockIdx.x
template <int K, int NO, bool ACT, bool BIAS>
__global__ void k_gemm(const float* __restrict__ A, const float* __restrict__ B,
                       const float* __restrict__ bias, float* __restrict__ C) {
    const int mt   = blockIdx.x;
    const int nt   = threadIdx.x >> 5;
    const int lane = threadIdx.x & 31;
    const int l    = lane & 15;
    const int half = lane >> 4;
    const int n0   = nt * 16;
    const float* Arow = A + (size_t)(mt * 16 + l) * K;   // A-frag: lane holds row M=l
    v8f acc = {0.f, 0.f, 0.f, 0.f, 0.f, 0.f, 0.f, 0.f};
#pragma unroll
    for (int ks = 0; ks < K / 4; ++ks) {
        const int k = ks * 4 + 2 * half;                 // K slots 2*half, 2*half+1
        v2f a = *(const v2f*)(Arow + k);                 // 8B-aligned (k even)
        v2f b;
        b.x = B[(size_t)k * NO + n0 + l];
        b.y = B[(size_t)(k + 1) * NO + n0 + l];
        acc = wmma4(a, b, acc);
    }
    const float bv = BIAS ? bias[n0 + l] : 0.f;
#pragma unroll
    for (int v = 0; v < 8; ++v) {                        // C: M = v + 8*half, N = l
        float val = acc[v] + bv;
        if (ACT) val = leaky(val);
        C[(size_t)(mt * 16 + v + 8 * half) * NO + n0 + l] = val;
    }
}

// ---------------- edge MLP: out[e] = leaky(h2[r]@Wfc[0:64] + h2[c]@Wfc[64:128] + bfc) ----
// grid = EE/16 edge tiles, blockDim = 64 (2 waves: N tiles 0..1 of 32 outputs)
__global__ void k_edge_mlp(const int* __restrict__ row, const int* __restrict__ col,
                           const float* __restrict__ h2, const float* __restrict__ Wfc,
                           const float* __restrict__ bfc, float* __restrict__ out) {
    const int et   = blockIdx.x;
    const int nt   = threadIdx.x >> 5;
    const int lane = threadIdx.x & 31;
    const int l    = lane & 15;
    const int half = lane >> 4;
    const int n0   = nt * 16;
    const int e    = et * 16 + l;                        // A-frag row M=l -> edge e
    const int r    = row[e];
    const int c    = col[e];
    const float* Ar = h2 + (size_t)r * 64;
    const float* Ac = h2 + (size_t)c * 64;
    v8f acc = {0.f, 0.f, 0.f, 0.f, 0.f, 0.f, 0.f, 0.f};
#pragma unroll
    for (int ks = 0; ks < 16; ++ks) {                    // K = 0..63 : source-node half
        const int k = ks * 4 + 2 * half;
        v2f a = *(const v2f*)(Ar + k);
        v2f b;
        b.x = Wfc[(size_t)k * 32 + n0 + l];
        b.y = Wfc[(size_t)(k + 1) * 32 + n0 + l];
        acc = wmma4(a, b, acc);
    }
#pragma unroll
    for (int ks = 0; ks < 16; ++ks) {                    // K = 64..127 : dest-node half
        const int k = ks * 4 + 2 * half;
        v2f a = *(const v2f*)(Ac + k);
        v2f b;
        b.x = Wfc[(size_t)(64 + k) * 32 + n0 + l];
        b.y = Wfc[(size_t)(64 + k + 1) * 32 + n0 + l];
        acc = wmma4(a, b, acc);
    }
    const float bv = bfc[n0 + l];
#pragma unroll
    for (int v = 0; v < 8; ++v) {
        float val = acc[v] + bv;
        out[(size_t)(et * 16 + v + 8 * half) * 32 + n0 + l] = leaky(val);
    }
}

extern "C" void kernel_launch(void* const* d_in, const int* in_sizes, int n_in,
                              void* d_out, int out_size, void* d_ws, size_t ws_size,
                              hipStream_t stream) {
    (void)in_sizes; (void)n_in; (void)out_size; (void)ws_size;
    const float* x   = (const float*)d_in[0];
    const int*   ei  = (const int*)d_in[1];          // [2, E] int32 (jax x64 disabled)
    const float* W1  = (const float*)d_in[2];        // [64,128]
    const float* b1  = (const float*)d_in[3];        // [128]
    const float* W2  = (const float*)d_in[4];        // [128,64]
    const float* b2  = (const float*)d_in[5];        // [64]
    const float* Wfc = (const float*)d_in[6];        // [128,32]
    const float* bfc = (const float*)d_in[7];        // [32]
    float* out = (float*)d_out;                      // [E,32]

    const int* row = ei;            // edge_index[0]
    const int* col = ei + EE;       // edge_index[1]

    float* ws   = (float*)d_ws;
    float* dinv = ws;                               // [N]       (deg -> dinv in place)
    float* agg0 = dinv + NN;                        // [N,64]
    float* h1   = agg0 + (size_t)NN * 64;           // [N,128]
    float* t2   = h1   + (size_t)NN * 128;          // [N,64]
    float* h2   = t2   + (size_t)NN * 64;           // [N,64]

    // --- normalization coefficients ---
    k_init_deg<<<(NN + 255) / 256, 256, 0, stream>>>(dinv);
    k_count_deg<<<(EE + 255) / 256, 256, 0, stream>>>(col, dinv);
    k_rsqrt<<<(NN + 255) / 256, 256, 0, stream>>>(dinv);

    // --- layer 1: aggregate x (64-dim), then h1 = leaky(agg0 @ W1 + b1) ---
    k_self_init<<<(NN * 64 + 255) / 256, 256, 0, stream>>>(x, dinv, agg0);
    k_scatter64<<<(EE * 64) / 256, 256, 0, stream>>>(row, col, dinv, x, agg0);
    k_gemm<64, 128, true, true><<<NN / 16, 256, 0, stream>>>(agg0, W1, b1, h1);

    // --- layer 2: t2 = h1 @ W2, aggregate (64-dim), h2 = leaky(agg + b2) ---
    k_gemm<128, 64, false, false><<<NN / 16, 128, 0, stream>>>(h1, W2, nullptr, t2);
    k_self_init<<<(NN * 64 + 255) / 256, 256, 0, stream>>>(t2, dinv, h2);
    k_scatter64<<<(EE * 64) / 256, 256, 0, stream>>>(row, col, dinv, t2, h2);
    k_bias_act64<<<(NN * 64 + 255) / 256, 256, 0, stream>>>(h2, b2);

    // --- edge MLP ---
    k_edge_mlp<<<EE / 16, 64, 0, stream>>>(row, col, h2, Wfc, bfc, out);
}